// LinearFeature_19387482374244
// MI455X (gfx1250) — compile-verified
//
#include <hip/hip_runtime.h>
#include <stdint.h>

// out[b,i,j,k] = x[b,i,j,k] * weight[i] + bias[i]
// B=8, N=512, M=128  -> 4096 slices of 16384 floats each.
// Roofline: 2 FLOP/elem, 8 B/elem round trip -> AI = 0.25 FLOP/B.
// Bound by 23.3 TB/s HBM (~23us floor); WMMA is irrelevant for this op.
// Data path: global_load_async_to_lds_b128 th:TH_LOAD_NT (ASYNCcnt, 4-deep
// pipeline) -> ds_load_b128 -> v_fma_f32 x4 -> global_store_b128 (NT).

typedef float v4f __attribute__((ext_vector_type(4)));

// GV mode: 64-bit vaddr pair, saddr=off. VDST holds the LDS byte address.
// The LDS offset is the low 32 bits of the flat shared-memory address
// (aperture layout: addr[31:0] is the wave-relative LDS offset).
// TH_LOAD_NT: one-touch stream, don't let 256MB of dead data rinse the L2.
__device__ __forceinline__ void async_copy_b128(const v4f* __restrict__ g, v4f* l) {
  unsigned lds_off = (unsigned)(uintptr_t)l;
  asm volatile("global_load_async_to_lds_b128 %0, %1, off th:TH_LOAD_NT"
               :: "v"(lds_off), "v"(g)
               : "memory");
}

// Constant-immediate waits; "memory" clobber stops the compiler from hoisting
// the LDS reads above the wait.
#define WAIT_ASYNC(n) asm volatile("s_wait_asynccnt " #n ::: "memory")

constexpr int kN        = 512;            // channels (power of two)
constexpr int kMM       = 128 * 128;      // elements per (b,i) slice
constexpr int kBlock    = 256;            // 8 waves (wave32)
constexpr int kV4Slice  = kMM / 4;        // 4096 float4 per slice
constexpr int kStages   = kV4Slice / kBlock; // 16 stages of 4KB per block
constexpr int kPipe     = 4;              // async pipeline depth (16KB LDS)

__global__ __launch_bounds__(kBlock)
void LinearFeature_scale_bias_stream(const float* __restrict__ x,
                                     const float* __restrict__ w,
                                     const float* __restrict__ b,
                                     float* __restrict__ out) {
  __shared__ v4f buf[kPipe][kBlock];

  const unsigned slice = blockIdx.x;        // b*N + i
  const unsigned ch    = slice & (kN - 1);  // channel i (uniform -> SGPR)
  const float scale = w[ch];                // s_load
  const float shift = b[ch];                // s_load

  const size_t base = (size_t)slice * (size_t)kMM;
  const v4f* __restrict__ src = (const v4f*)(x + base);
  v4f* __restrict__ dst = (v4f*)(out + base);
  const unsigned tid = threadIdx.x;

  auto consume = [&](int st) {
    v4f v = buf[st & (kPipe - 1)][tid];     // ds_load_b128 (own lane's slot)
    v4f r;
    r.x = fmaf(v.x, scale, shift);
    r.y = fmaf(v.y, scale, shift);
    r.z = fmaf(v.z, scale, shift);
    r.w = fmaf(v.w, scale, shift);
    // dead stream: NT hint so the outbound 256MB doesn't rinse the 192MB L2
    __builtin_nontemporal_store(r, dst + (size_t)st * kBlock + tid);
  };

  // Prologue: fill pipeline (3 stages in flight).
  async_copy_b128(src + 0 * kBlock + tid, &buf[0][tid]);
  async_copy_b128(src + 1 * kBlock + tid, &buf[1][tid]);
  async_copy_b128(src + 2 * kBlock + tid, &buf[2][tid]);

  // Steady state: issue st+3, wait until <=3 outstanding (=> stage st done,
  // async loads complete in order per wave), consume st. Each lane only reads
  // the LDS slot it filled itself, so per-wave ASYNCcnt ordering suffices —
  // no barriers needed.
#pragma unroll 1
  for (int st = 0; st < kStages - (kPipe - 1); ++st) {
    async_copy_b128(src + (size_t)(st + kPipe - 1) * kBlock + tid,
                    &buf[(st + kPipe - 1) & (kPipe - 1)][tid]);
    WAIT_ASYNC(0x3);
    consume(st);
  }
  // Drain.
  WAIT_ASYNC(0x2); consume(kStages - 3);
  WAIT_ASYNC(0x1); consume(kStages - 2);
  WAIT_ASYNC(0x0); consume(kStages - 1);
}

extern "C" void kernel_launch(void* const* d_in, const int* in_sizes, int n_in,
                              void* d_out, int out_size, void* d_ws, size_t ws_size,
                              hipStream_t stream) {
  const float* x  = (const float*)d_in[0];   // (B, N, M, M) fp32
  const float* wt = (const float*)d_in[1];   // (N,) fp32
  const float* bs = (const float*)d_in[2];   // (N,) fp32
  float* out = (float*)d_out;

  const int slices = out_size / kMM;         // B*N = 4096
  dim3 grid((unsigned)slices);
  dim3 block(kBlock);
  hipLaunchKernelGGL(LinearFeature_scale_bias_stream, grid, block, 0, stream,
                     x, wt, bs, out);
  (void)in_sizes; (void)n_in; (void)d_ws; (void)ws_size;
}